// SSINF3OneTenthMLP_19198503813134
// MI455X (gfx1250) — compile-verified
//
#include <hip/hip_runtime.h>
#include <hip/hip_bf16.h>

typedef __attribute__((ext_vector_type(2))) float v2f;
typedef __attribute__((ext_vector_type(8))) float v8f;

#define NTOK 4096      // B*T
#define DIM  768       // D
#define SS   8         // subspaces
#define RR   16        // subspace rank
#define MMA  64        // splanifolds
#define TOPK 4
#define HH   32        // local hidden
#define GG   256       // global hidden
#define SR   (SS*RR)   // 128

__device__ __forceinline__ float gelu_tanh(float x) {
    // jax.nn.gelu default (approximate=True): 0.5*x*(1+tanh(sqrt(2/pi)*(x+0.044715*x^3)))
    const float c0 = 0.7978845608028654f;
    const float c1 = 0.044715f;
    float x3 = x * x * x;
    return 0.5f * x * (1.0f + tanhf(c0 * (x + c1 * x3)));
}

// ---------------------------------------------------------------------------
// Kernel 0: permute P [S, D, R] -> Pt [D, S*R] (row-major K x N for the GEMM)
// ---------------------------------------------------------------------------
__global__ void permuteP_kernel(const float* __restrict__ P, float* __restrict__ Pt) {
    int t = blockIdx.x * blockDim.x + threadIdx.x;
    if (t >= DIM * SR) return;
    int d = t / SR;
    int c = t - d * SR;
    int s = c >> 4;          // c / R
    int r = c & 15;          // c % R
    Pt[t] = P[((size_t)s * DIM + d) * RR + r];
}

// ---------------------------------------------------------------------------
// fp32 WMMA GEMM, 16x64 super-tile per wave (4 accumulators, A-fragment reuse)
// with software double-buffering: fragments for k+4 are prefetched before the
// WMMA group for k issues, so the loadcnt wait before each WMMA group drains
// only to the in-flight next-iteration loads instead of to zero.
// C[Nrows,Ncols] = act(A[Nrows,Kdim] @ B[Kdim,Ncols] + bias)
// Operand layouts per CDNA5 ISA 7.12.2 (32-bit A 16x4 / B 4x16 / C,D 16x16).
// Requires: Ncols % 64 == 0, Kdim % 4 == 0 and Kdim >= 8, grid exact.
// ---------------------------------------------------------------------------
__global__ void gemm_f32_wmma4_kernel(const float* __restrict__ A,
                                      const float* __restrict__ Bm,
                                      const float* __restrict__ bias,
                                      float* __restrict__ Cm,
                                      int Nrows, int Kdim, int Ncols,
                                      int applyGelu) {
    const int lane = threadIdx.x & 31;
    const int wave = threadIdx.x >> 5;
    const int tile = blockIdx.x * (blockDim.x >> 5) + wave;   // one 16x64 super-tile
    const int tcols = Ncols >> 6;
    const int trow = tile / tcols;
    const int tcol = tile - trow * tcols;
    if (trow * 16 >= Nrows) return;             // wave-uniform guard (EXEC stays all-1s)

    const int row0 = trow << 4;
    const int col0 = tcol << 6;
    const int m    = lane & 15;                 // M for A lanes / N for B,C,D lanes
    const int kh   = (lane >> 4) << 1;          // 0 for lanes 0-15, 2 for lanes 16-31

    v8f acc[4];
    #pragma unroll
    for (int c = 0; c < 4; ++c)
        acc[c] = (v8f){0.f, 0.f, 0.f, 0.f, 0.f, 0.f, 0.f, 0.f};

    const float* ap = A  + (size_t)(row0 + m) * Kdim + kh;
    const float* bp = Bm + (size_t)kh * Ncols + col0 + m;

    // prologue: load fragments for k = 0
    v2f a0 = *(const v2f*)ap;                   // A[row0+m, kh .. kh+1], 8B aligned
    v2f b0[4];
    #pragma unroll
    for (int c = 0; c < 4; ++c) {
        b0[c][0] = bp[c * 16];                  // B[kh,   col0+16c+m]
        b0[c][1] = bp[c * 16 + Ncols];          // B[kh+1, col0+16c+m]
    }

    // steady state: prefetch k+4, then consume k
    for (int k0 = 0; k0 < Kdim - 4; k0 += 4) {
        ap += 4;
        bp += 4 * (size_t)Ncols;
        v2f a1 = *(const v2f*)ap;
        v2f b1[4];
        #pragma unroll
        for (int c = 0; c < 4; ++c) {
            b1[c][0] = bp[c * 16];
            b1[c][1] = bp[c * 16 + Ncols];
        }
        #pragma unroll
        for (int c = 0; c < 4; ++c) {
            acc[c] = __builtin_amdgcn_wmma_f32_16x16x4_f32(
                /*neg_a=*/false, a0, /*neg_b=*/false, b0[c],
                /*c_mod=*/(short)0, acc[c], /*reuse_a=*/false, /*reuse_b=*/false);
        }
        a0 = a1;
        #pragma unroll
        for (int c = 0; c < 4; ++c) b0[c] = b1[c];
    }

    // epilogue: consume the final fragments
    #pragma unroll
    for (int c = 0; c < 4; ++c) {
        acc[c] = __builtin_amdgcn_wmma_f32_16x16x4_f32(
            false, a0, false, b0[c], (short)0, acc[c], false, false);
    }

    const int rbase = row0 + ((lane >> 4) << 3);   // +8 rows for upper half lanes
    #pragma unroll
    for (int c = 0; c < 4; ++c) {
        const int col = col0 + c * 16 + m;
        const float bval = bias ? bias[col] : 0.0f;
        #pragma unroll
        for (int j = 0; j < 8; ++j) {
            float v = acc[c][j] + bval;
            if (applyGelu) v = gelu_tanh(v);
            Cm[(size_t)(rbase + j) * Ncols + col] = v;
        }
    }
}

// ---------------------------------------------------------------------------
// Kernel 2: routing + per-splanifold expert MLPs.
// One thread per (token n, subspace s). z row (16 floats) in registers,
// 64 anchor scores -> explicit top-4 -> softmax -> 4 experts of
// gelu(z@W1[16x32]+b1)@W2[32x16]+b2 -> weighted mix -> comb[n, s*16..].
// Expert weights (2 MB total) are L2-resident, so gathers are cheap.
// ---------------------------------------------------------------------------
__global__ void experts_kernel(const float* __restrict__ z,
                               const float* __restrict__ C,
                               const float* __restrict__ W1,
                               const float* __restrict__ b1,
                               const float* __restrict__ W2,
                               const float* __restrict__ b2,
                               float* __restrict__ comb) {
    int t = blockIdx.x * blockDim.x + threadIdx.x;
    if (t >= NTOK * SS) return;
    int n = t / SS;
    int s = t - n * SS;

    // load z[n, s, :]
    float zr[RR];
    const float* zp = z + (size_t)n * SR + s * RR;
    #pragma unroll
    for (int r = 0; r < RR; ++r) zr[r] = zp[r];

    // scores over M anchors, maintain sorted top-4 (ties -> lower index wins)
    float tv0 = -3.4e38f, tv1 = -3.4e38f, tv2 = -3.4e38f, tv3 = -3.4e38f;
    int   ti0 = 0, ti1 = 0, ti2 = 0, ti3 = 0;
    const float* Cs = C + (size_t)s * MMA * RR;
    for (int mm = 0; mm < MMA; ++mm) {
        const float* cp = Cs + mm * RR;
        float sc = 0.0f;
        #pragma unroll
        for (int r = 0; r < RR; ++r) sc += zr[r] * cp[r];
        if (sc > tv0)      { tv3=tv2; ti3=ti2; tv2=tv1; ti2=ti1; tv1=tv0; ti1=ti0; tv0=sc; ti0=mm; }
        else if (sc > tv1) { tv3=tv2; ti3=ti2; tv2=tv1; ti2=ti1; tv1=sc; ti1=mm; }
        else if (sc > tv2) { tv3=tv2; ti3=ti2; tv2=sc; ti2=mm; }
        else if (sc > tv3) { tv3=sc; ti3=mm; }
    }

    // softmax over the 4 selected scores
    float e0 = 1.0f;                   // expf(tv0 - tv0)
    float e1 = expf(tv1 - tv0);
    float e2 = expf(tv2 - tv0);
    float e3 = expf(tv3 - tv0);
    float inv = 1.0f / (e0 + e1 + e2 + e3);
    float wk[TOPK] = {e0 * inv, e1 * inv, e2 * inv, e3 * inv};
    int   mi[TOPK] = {ti0, ti1, ti2, ti3};

    float accv[RR];
    #pragma unroll
    for (int r = 0; r < RR; ++r) accv[r] = 0.0f;

    for (int k = 0; k < TOPK; ++k) {
        const int    mm  = mi[k];
        const float  wgt = wk[k];
        const size_t e   = (size_t)s * MMA + mm;
        const float* w1p = W1 + e * RR * HH;   // [R, H] row-major
        const float* b1p = b1 + e * HH;
        const float* w2p = W2 + e * HH * RR;   // [H, R] row-major
        const float* b2p = b2 + e * RR;

        float y[RR];
        #pragma unroll
        for (int r = 0; r < RR; ++r) y[r] = b2p[r];

        for (int j = 0; j < HH; ++j) {         // rolled: hv stays scalar
            float hv = b1p[j];
            #pragma unroll
            for (int r = 0; r < RR; ++r) hv += zr[r] * w1p[r * HH + j];
            hv = gelu_tanh(hv);
            const float* row2 = w2p + j * RR;
            #pragma unroll
            for (int r = 0; r < RR; ++r) y[r] += hv * row2[r];
        }
        #pragma unroll
        for (int r = 0; r < RR; ++r) accv[r] += wgt * y[r];
    }

    float* op = comb + (size_t)n * SR + s * RR;
    #pragma unroll
    for (int r = 0; r < RR; ++r) op[r] = accv[r];
}

// ---------------------------------------------------------------------------
// Launcher: x,P,C,W1,b1,W2,b2,G1,g1b,G2,g2b (all f32). out = [B,T,D] f32.
// ---------------------------------------------------------------------------
extern "C" void kernel_launch(void* const* d_in, const int* in_sizes, int n_in,
                              void* d_out, int out_size, void* d_ws, size_t ws_size,
                              hipStream_t stream) {
    const float* x   = (const float*)d_in[0];
    const float* P   = (const float*)d_in[1];
    const float* C   = (const float*)d_in[2];
    const float* W1  = (const float*)d_in[3];
    const float* b1  = (const float*)d_in[4];
    const float* W2  = (const float*)d_in[5];
    const float* b2  = (const float*)d_in[6];
    const float* G1  = (const float*)d_in[7];
    const float* g1b = (const float*)d_in[8];
    const float* G2  = (const float*)d_in[9];
    const float* g2b = (const float*)d_in[10];
    float* out = (float*)d_out;

    // workspace layout (floats), ~8.8 MB total
    float* ws   = (float*)d_ws;
    float* Pt   = ws;                       // DIM * SR            =  98304
    float* z    = Pt + (size_t)DIM * SR;    // NTOK * SR           = 524288
    float* comb = z  + (size_t)NTOK * SR;   // NTOK * SR           = 524288
    float* hid  = comb + (size_t)NTOK * SR; // NTOK * GG           = 1048576

    // k0: P -> Pt
    permuteP_kernel<<<(DIM * SR + 255) / 256, 256, 0, stream>>>(P, Pt);

    // k1: z = x @ Pt          [4096 x 768] x [768 x 128]
    {
        int tiles = (NTOK / 16) * (SR / 64);      // 512 super-tiles
        gemm_f32_wmma4_kernel<<<tiles / 8, 256, 0, stream>>>(
            x, Pt, nullptr, z, NTOK, DIM, SR, 0);
    }

    // k2: routing + experts -> comb [4096 x 128]
    experts_kernel<<<(NTOK * SS) / 256, 256, 0, stream>>>(z, C, W1, b1, W2, b2, comb);

    // k3: hid = gelu(comb @ G1 + g1b)   [4096 x 128] x [128 x 256]
    {
        int tiles = (NTOK / 16) * (GG / 64);      // 1024 super-tiles
        gemm_f32_wmma4_kernel<<<tiles / 8, 256, 0, stream>>>(
            comb, G1, g1b, hid, NTOK, SR, GG, 1);
    }

    // k4: out = hid @ G2 + g2b          [4096 x 256] x [256 x 768]
    {
        int tiles = (NTOK / 16) * (DIM / 64);     // 3072 super-tiles
        gemm_f32_wmma4_kernel<<<tiles / 8, 256, 0, stream>>>(
            hid, G2, g2b, out, NTOK, GG, DIM, 0);
    }
}